// GaussianField_61263413510370
// MI455X (gfx1250) — compile-verified
//
#include <hip/hip_runtime.h>
#include <math.h>

// CDNA5 (gfx1250) wave32 WMMA types
typedef float v2f __attribute__((ext_vector_type(2)));
typedef float v8f __attribute__((ext_vector_type(8)));
typedef float v4f __attribute__((ext_vector_type(4)));

// Fast sigmoid: v_exp_f32 + v_rcp_f32 (avoids the IEEE div_scale/div_fixup
// expansion; ~1ulp on rcp is irrelevant through a sigmoid).
__device__ __forceinline__ float sigmoid_fast(float v) {
    return __builtin_amdgcn_rcpf(1.0f + __expf(-v));
}

// ---------------------------------------------------------------------------
// Colors kernel: one gaussian per wave-iteration.
//   D[b,c] = sum_k basis[b,k] * sh[n,c,k],  K=9 padded to 12 -> 3x WMMA
//   V_WMMA_F32_16X16X4_F32 chained on the f32 accumulator.
//
// A-matrix 16x4 f32 layout (ISA 7.12.2): VGPR0 holds K=kb (lanes 0-15) and
// K=kb+2 (lanes 16-31); VGPR1 holds kb+1 / kb+3. B 4x16 mirrored. D: VGPR r,
// lane c.
//
// No cross-lane traffic is needed for A: lane l must supply basis row
// m = l&15, and lane l computes the basis for batch b = l&7. For all rows
// that matter (m < 8, since D rows 8..15 are discarded) we have m == b, so
// the lane's own registers already hold the right row -> pure v_cndmask
// fragment construction, EXEC stays all-ones through the WMMAs.
//
// Normalization: 1/max(||v||,1e-12) == rsqrt(max(dot,1e-24)) exactly
// (sqrt is monotone), so the IEEE sqrt+div expansion collapses to
// v_max_num_f32 + v_rsq_f32.
// ---------------------------------------------------------------------------
__global__ void gf_colors_wmma(const float* __restrict__ vdirs,  // [B,N,3]
                               const float* __restrict__ sh,     // [N,3,9]
                               float* __restrict__ colors,       // [B,N,3]
                               long long N, int B) {
    const int lane = threadIdx.x & 31;
    const int waveInBlk = threadIdx.x >> 5;
    const int wavesPerBlk = blockDim.x >> 5;
    const long long wave = (long long)blockIdx.x * wavesPerBlk + waveInBlk;
    const long long nWaves = (long long)gridDim.x * wavesPerBlk;

    const int b = lane & 7;        // batch whose basis this lane computes
    const int c = lane & 15;       // B/D column (cols 3..15 are padding)
    const bool hi = (lane >= 16);  // upper half-wave supplies K = kb+2, kb+3

    for (long long n = wave; n < N; n += nWaves) {
        // ---- per-lane SH basis for batch b ------------------------------
        const float* vp = vdirs + ((long long)b * N + n) * 3;
        float x = vp[0], y = vp[1], z = vp[2];
        float dot = x * x + y * y + z * z;
        float inv = __builtin_amdgcn_rsqf(fmaxf(dot, 1e-24f));
        x *= inv; y *= inv; z *= inv;
        const float bas0 = 1.0f, bas1 = x, bas2 = y, bas3 = z;
        const float bas4 = x * y, bas5 = x * z, bas6 = y * z;
        const float bas7 = x * x, bas8 = y * y;   // (zz dropped: K=9)

        // ---- B operand: lane's channel row sh[n, c, 0..8]; zero pad c>=3
        float s0 = 0.f, s1 = 0.f, s2 = 0.f, s3 = 0.f, s4 = 0.f,
              s5 = 0.f, s6 = 0.f, s7 = 0.f, s8 = 0.f;
        if (c < 3) {
            const float* p = sh + (long long)n * 27 + c * 9;
            s0 = p[0]; s1 = p[1]; s2 = p[2]; s3 = p[3]; s4 = p[4];
            s5 = p[5]; s6 = p[6]; s7 = p[7]; s8 = p[8];
        }

        v8f acc = {};

        // ---- K-tile 0: k = 0..3 ----------------------------------------
        {
            v2f a, bf;
            a[0]  = hi ? bas2 : bas0;
            a[1]  = hi ? bas3 : bas1;
            bf[0] = hi ? s2 : s0;
            bf[1] = hi ? s3 : s1;
            acc = __builtin_amdgcn_wmma_f32_16x16x4_f32(
                false, a, false, bf, (short)0, acc, false, false);
        }
        // ---- K-tile 1: k = 4..7 ----------------------------------------
        {
            v2f a, bf;
            a[0]  = hi ? bas6 : bas4;
            a[1]  = hi ? bas7 : bas5;
            bf[0] = hi ? s6 : s4;
            bf[1] = hi ? s7 : s5;
            acc = __builtin_amdgcn_wmma_f32_16x16x4_f32(
                false, a, false, bf, (short)0, acc, false, false);
        }
        // ---- K-tile 2: k = 8 (9..11 are zero padding) ------------------
        {
            v2f a, bf;
            a[0]  = hi ? 0.f : bas8;
            a[1]  = 0.f;
            bf[0] = hi ? 0.f : s8;
            bf[1] = 0.f;
            acc = __builtin_amdgcn_wmma_f32_16x16x4_f32(
                false, a, false, bf, (short)0, acc, false, false);
        }

        // ---- D[r][c] lives in acc[r] of lane c (rows 0..7 = batches) ---
        // B is wave-uniform: branch once, not per unrolled element.
        if (lane < 3) {
            if (B == 8) {
#pragma unroll
                for (int r = 0; r < 8; ++r) {
                    colors[((long long)r * N + n) * 3 + lane] =
                        sigmoid_fast(acc[r]);
                }
            } else {
                const int rmax = B < 8 ? B : 8;
                for (int r = 0; r < rmax; ++r) {
                    colors[((long long)r * N + n) * 3 + lane] =
                        sigmoid_fast(acc[r]);
                }
            }
        }
    }
}

// ---------------------------------------------------------------------------
// Pass-through outputs (bandwidth bound): out layout is the flat concat
//   [0,3N) means | [3N,6N) exp(log_scales) | [6N,10N) rotations
//   [10N,37N) sh_coeffs | [37N,38N) opacities
// Vectorized float4 path (all boundaries are multiples of 4 when N%4==0).
// ---------------------------------------------------------------------------
__global__ void gf_pass_v4(const float* __restrict__ means,
                           const float* __restrict__ logs,
                           const float* __restrict__ rots,
                           const float* __restrict__ sh,
                           const float* __restrict__ opa,
                           float* __restrict__ out, long long N) {
    const long long q3 = 3 * N / 4, q6 = 6 * N / 4, q10 = 10 * N / 4,
                    q37 = 37 * N / 4, q38 = 38 * N / 4;
    for (long long i = (long long)blockIdx.x * blockDim.x + threadIdx.x; i < q38;
         i += (long long)gridDim.x * blockDim.x) {
        v4f v;
        if (i < q3) {
            v = ((const v4f*)means)[i];
        } else if (i < q6) {
            v4f l = ((const v4f*)logs)[i - q3];
            v[0] = expf(l[0]); v[1] = expf(l[1]);
            v[2] = expf(l[2]); v[3] = expf(l[3]);
        } else if (i < q10) {
            v = ((const v4f*)rots)[i - q6];
        } else if (i < q37) {
            v = ((const v4f*)sh)[i - q10];
        } else {
            v = ((const v4f*)opa)[i - q37];
        }
        ((v4f*)out)[i] = v;
    }
}

__global__ void gf_pass_scalar(const float* __restrict__ means,
                               const float* __restrict__ logs,
                               const float* __restrict__ rots,
                               const float* __restrict__ sh,
                               const float* __restrict__ opa,
                               float* __restrict__ out, long long N) {
    const long long n3 = 3 * N, n6 = 6 * N, n10 = 10 * N, n37 = 37 * N,
                    n38 = 38 * N;
    for (long long i = (long long)blockIdx.x * blockDim.x + threadIdx.x; i < n38;
         i += (long long)gridDim.x * blockDim.x) {
        float v;
        if (i < n3)       v = means[i];
        else if (i < n6)  v = expf(logs[i - n3]);
        else if (i < n10) v = rots[i - n6];
        else if (i < n37) v = sh[i - n10];
        else              v = opa[i - n37];
        out[i] = v;
    }
}

extern "C" void kernel_launch(void* const* d_in, const int* in_sizes, int n_in,
                              void* d_out, int out_size, void* d_ws, size_t ws_size,
                              hipStream_t stream) {
    const float* vdirs = (const float*)d_in[0];  // [B,N,3]
    const float* means = (const float*)d_in[1];  // [N,3]
    const float* logs  = (const float*)d_in[2];  // [N,3]
    const float* rots  = (const float*)d_in[3];  // [N,4]
    const float* sh    = (const float*)d_in[4];  // [N,3,9]
    const float* opa   = (const float*)d_in[5];  // [N,1]
    float* out = (float*)d_out;

    const long long N = in_sizes[1] / 3;
    const int B = (int)((long long)in_sizes[0] / (3 * N));
    float* colors = out + 38LL * N;

    if ((N & 3) == 0) {
        gf_pass_v4<<<2048, 256, 0, stream>>>(means, logs, rots, sh, opa, out, N);
    } else {
        gf_pass_scalar<<<2048, 256, 0, stream>>>(means, logs, rots, sh, opa, out, N);
    }
    gf_colors_wmma<<<2048, 256, 0, stream>>>(vdirs, sh, colors, N, B);
}